// SpectralAttentionNet_65111704207557
// MI455X (gfx1250) — compile-verified
//
#include <hip/hip_runtime.h>
#include <math.h>

#define NSAMP   2048
#define NPTS    10000
#define NBATCH  4
#define KMAX    200
#define FD      64
#define NFEAT   128
#define NLVL    4

typedef __attribute__((ext_vector_type(2))) float v2f;
typedef __attribute__((ext_vector_type(8))) float v8f;

// ---------------------------------------------------------------------------
// 1) Farthest point sampling: one workgroup per batch, points resident in
//    registers (10/thread), LDS only for the 1024-wide argmax tree.
// ---------------------------------------------------------------------------
__global__ __launch_bounds__(1024) void fps_kernel(const float* __restrict__ xyz,
                                                   int* __restrict__ idx) {
  const int b = blockIdx.x;
  const int t = threadIdx.x;
  const float* base = xyz + (size_t)b * NPTS * 3;

  float px[10], py[10], pz[10], md[10];
#pragma unroll
  for (int q = 0; q < 10; ++q) {
    int pp = t + q * 1024;
    if (pp < NPTS) {
      px[q] = base[pp * 3 + 0];
      py[q] = base[pp * 3 + 1];
      pz[q] = base[pp * 3 + 2];
    } else { px[q] = 0.f; py[q] = 0.f; pz[q] = 0.f; }
    md[q] = 1e30f;
  }

  __shared__ float rv[1024];
  __shared__ int   ri[1024];

  int cur = 0;
  if (t == 0) idx[b * NSAMP] = 0;

  for (int s = 1; s < NSAMP; ++s) {
    float cx = base[cur * 3 + 0];
    float cy = base[cur * 3 + 1];
    float cz = base[cur * 3 + 2];
    float bv = -1.0f; int bi = NPTS;
#pragma unroll
    for (int q = 0; q < 10; ++q) {
      int pp = t + q * 1024;
      if (pp < NPTS) {
        float dx = px[q] - cx, dy = py[q] - cy, dz = pz[q] - cz;
        float d = dx * dx + dy * dy + dz * dz;
        if (d < md[q]) md[q] = d;
        if (md[q] > bv || (md[q] == bv && pp < bi)) { bv = md[q]; bi = pp; }
      }
    }
    rv[t] = bv; ri[t] = bi;
    __syncthreads();
    for (int off = 512; off > 0; off >>= 1) {
      if (t < off) {
        float ov = rv[t + off]; int oi = ri[t + off];
        if (ov > rv[t] || (ov == rv[t] && oi < ri[t])) { rv[t] = ov; ri[t] = oi; }
      }
      __syncthreads();
    }
    cur = ri[0];
    if (t == 0) idx[b * NSAMP + s] = cur;
    __syncthreads();
  }
}

// ---------------------------------------------------------------------------
// 2) Gather sub-sampled eigenvectors + masses
// ---------------------------------------------------------------------------
__global__ void gather_kernel(const float* __restrict__ ev0, const float* __restrict__ ev1,
                              const float* __restrict__ mass1,
                              const int* __restrict__ idx0, const int* __restrict__ idx1,
                              float* __restrict__ ev0s, float* __restrict__ ev1s,
                              float* __restrict__ mass1s) {
  int tid = blockIdx.x * blockDim.x + threadIdx.x;
  const int total = NBATCH * NSAMP * KMAX;
  if (tid >= total) return;
  int k = tid % KMAX;
  int s = (tid / KMAX) % NSAMP;
  int b = tid / (KMAX * NSAMP);
  int i0 = idx0[b * NSAMP + s];
  int i1 = idx1[b * NSAMP + s];
  ev0s[tid] = ev0[((size_t)b * NPTS + i0) * KMAX + k];
  ev1s[tid] = ev1[((size_t)b * NPTS + i1) * KMAX + k];
  if (k == 0) mass1s[b * NSAMP + s] = mass1[(size_t)b * NPTS + i1];
}

// ---------------------------------------------------------------------------
// 3a) e0 = evecs0_sub[:, :, :k] @ fm^T  (per level), written with padded row
//     stride k1p (pad lanes zeroed) so the WMMA kernel needs no K masking.
// ---------------------------------------------------------------------------
__global__ void e0_gemm_kernel(const float* __restrict__ ev0s, const float* __restrict__ fm,
                               float* __restrict__ e0, int k1, int k1p) {
  int tid = blockIdx.x * blockDim.x + threadIdx.x;
  int total = NBATCH * NSAMP * k1p;
  if (tid >= total) return;
  int j = tid % k1p;
  int m = (tid / k1p) % NSAMP;
  int b = tid / (k1p * NSAMP);
  float acc = 0.f;
  if (j < k1) {
    const float* xr = ev0s + ((size_t)b * NSAMP + m) * KMAX;
    const float* fr = fm + ((size_t)b * k1 + j) * k1;
    for (int k = 0; k < k1; ++k) acc += xr[k] * fr[k];
  }
  e0[tid] = acc;   // zero in the padding lanes
}

// 3b) squared norms of e1 rows (first k1 comps) and e0 rows (stride k1p)
__global__ void norms_kernel(const float* __restrict__ ev1s, const float* __restrict__ e0,
                             float* __restrict__ n1, float* __restrict__ n0,
                             int k1, int k1p) {
  int tid = blockIdx.x * blockDim.x + threadIdx.x;
  if (tid >= NBATCH * NSAMP) return;
  const float* r1 = ev1s + (size_t)tid * KMAX;
  float a = 0.f;
  for (int k = 0; k < k1; ++k) a += r1[k] * r1[k];
  n1[tid] = a;
  const float* r0 = e0 + (size_t)tid * k1p;
  float c = 0.f;
  for (int k = 0; k < k1; ++k) c += r0[k] * r0[k];
  n0[tid] = c;
}

// ---------------------------------------------------------------------------
// 3c) min_j ||e1_i - e0_j|| via f32 WMMA (V_WMMA_F32_16X16X4_F32).
//     K1P is a compile-time multiple of 4; K loop fully unrolled.  The j-tile
//     loop is unrolled x2 so each K step feeds TWO independent WMMA chains
//     (shared register-resident A fragment, two B streams) -> loads of one
//     chain overlap the other chain's WMMAs instead of serializing on
//     s_wait_loadcnt 0.  EXEC is all-1s throughout the matrix region.
//     C-tile layout (32-bit, 16x16): lanes 0-15 -> M=r, N=lane;
//                                    lanes 16-31 -> M=r+8, N=lane-16.
//     A (16x4 f32): vgpr v, lane l -> M=l%16, K = v + 2*(l>=16); B symmetric.
// ---------------------------------------------------------------------------
template <int K1, int K1P>
__global__ __launch_bounds__(256) void pd_min_wmma_kernel(
    const float* __restrict__ ev1s, const float* __restrict__ e0,
    const float* __restrict__ n1, const float* __restrict__ n0,
    float* __restrict__ res, float scale, int lvl) {
  const int strip = blockIdx.x;           // 0..127  (i-tile)
  const int b     = blockIdx.y;           // 0..3
  const int i0    = strip * 16;
  const int tid   = threadIdx.x;
  const int lane  = tid & 31;
  const int wave  = tid >> 5;             // 0..7
  const int half  = (lane >> 4) & 1;
  const int lcol  = lane & 15;
  const int rowoff = half * 8;

  __shared__ float s_e1[16 * K1P];
  __shared__ float s_n1[16];
  __shared__ float s_pmin[8 * 16 * 16];

  for (int t = tid; t < 16 * K1P; t += 256) {
    int r = t / K1P, k = t - r * K1P;
    s_e1[t] = (k < K1) ? ev1s[((size_t)b * NSAMP + i0 + r) * KMAX + k] : 0.0f;
  }
  if (tid < 16) s_n1[tid] = n1[b * NSAMP + i0 + tid];
  __syncthreads();

  float vm[8];
#pragma unroll
  for (int r = 0; r < 8; ++r) vm[r] = 3.0e38f;

  const float* s_arow = s_e1 + lcol * K1P + 2 * half;
  const float* e0base = e0 + ((size_t)b * NSAMP + lcol) * K1P + 2 * half;
  const float* n0base = n0 + b * NSAMP;

  // 128 j-tiles, 8 waves, 2 tiles per iteration -> 8 iterations, no remainder
  for (int it = 0; it < (NSAMP / 16) / 16; ++it) {
    const int jA = (wave + it * 16) * 16;       // tile 1 column base
    const int jB = jA + 8 * 16;                 // tile 2 column base
    const float* e0colA = e0base + (size_t)jA * K1P;
    const float* e0colB = e0base + (size_t)jB * K1P;

    v8f accA = {0.f, 0.f, 0.f, 0.f, 0.f, 0.f, 0.f, 0.f};
    v8f accB = {0.f, 0.f, 0.f, 0.f, 0.f, 0.f, 0.f, 0.f};
#pragma unroll
    for (int kc = 0; kc < K1P; kc += 4) {
      v2f a  = *(const v2f*)(s_arow + kc);     // 8B LDS load (shared A frag)
      v2f b0 = *(const v2f*)(e0colA + kc);     // 8B global load, chain A
      v2f b1 = *(const v2f*)(e0colB + kc);     // 8B global load, chain B
      accA = __builtin_amdgcn_wmma_f32_16x16x4_f32(false, a, false, b0,
                                                   (short)0, accA, false, false);
      accB = __builtin_amdgcn_wmma_f32_16x16x4_f32(false, a, false, b1,
                                                   (short)0, accB, false, false);
    }
    float n0a = n0base[jA + lcol];
    float n0b = n0base[jB + lcol];
#pragma unroll
    for (int r = 0; r < 8; ++r) {
      float n1r = s_n1[r + rowoff];
      float dA = fmaxf(n1r + n0a - 2.0f * accA[r], 0.0f);
      float dB = fmaxf(n1r + n0b - 2.0f * accB[r], 0.0f);
      vm[r] = fminf(vm[r], fminf(dA, dB));
    }
  }

#pragma unroll
  for (int r = 0; r < 8; ++r)
    s_pmin[wave * 256 + (r + rowoff) * 16 + lcol] = vm[r];
  __syncthreads();

  {   // min across waves (each thread owns one (row,col) slot)
    int row = tid >> 4, col = tid & 15;
    float m = s_pmin[row * 16 + col];
#pragma unroll
    for (int w = 1; w < 8; ++w) m = fminf(m, s_pmin[w * 256 + row * 16 + col]);
    s_pmin[row * 16 + col] = m;
  }
  __syncthreads();
  if (tid < 16) {   // min across the 16 column slots -> full row min
    float m = s_pmin[tid * 16];
    for (int c = 1; c < 16; ++c) m = fminf(m, s_pmin[tid * 16 + c]);
    res[((size_t)b * NLVL + lvl) * NSAMP + i0 + tid] = sqrtf(m) * scale;
  }
}

// ---------------------------------------------------------------------------
// 4) Network head kernels (tiny: ~0.5 GFLOP total)
// ---------------------------------------------------------------------------
__global__ void conv_bn_kernel(const float* __restrict__ x, const float* __restrict__ w,
                               const float* __restrict__ bias, const float* __restrict__ gamma,
                               const float* __restrict__ beta, const float* __restrict__ mean,
                               const float* __restrict__ var, float* __restrict__ y,
                               int Cin, int Cout, int act) {
  int tid = blockIdx.x * blockDim.x + threadIdx.x;
  int total = NBATCH * Cout * NSAMP;
  if (tid >= total) return;
  int n = tid % NSAMP;
  int o = (tid / NSAMP) % Cout;
  int b = tid / (NSAMP * Cout);
  const float* xb = x + (size_t)b * Cin * NSAMP + n;
  const float* wr = w + (size_t)o * Cin;
  float acc = bias[o];
  for (int c = 0; c < Cin; ++c) acc += wr[c] * xb[(size_t)c * NSAMP];
  float s = gamma[o] / sqrtf(var[o] + 1e-5f);
  float v = (acc - mean[o]) * s + beta[o];
  y[tid] = act ? fmaxf(v, 0.f) : v;
}

__global__ __launch_bounds__(256) void maxpool_kernel(const float* __restrict__ x,
                                                      float* __restrict__ y) {
  int bo = blockIdx.x;                 // b*FD + o
  int t = threadIdx.x;
  const float* xr = x + (size_t)bo * NSAMP;
  float m = -3e38f;
  for (int n = t; n < NSAMP; n += 256) m = fmaxf(m, xr[n]);
  __shared__ float sm[256];
  sm[t] = m; __syncthreads();
  for (int off = 128; off > 0; off >>= 1) {
    if (t < off) sm[t] = fmaxf(sm[t], sm[t + off]);
    __syncthreads();
  }
  if (t == 0) y[bo] = sm[0];
}

__global__ void linear_kernel(const float* __restrict__ x, const float* __restrict__ w,
                              const float* __restrict__ bias, float* __restrict__ y,
                              int Cin, int Cout, int act) {
  int tid = blockIdx.x * blockDim.x + threadIdx.x;
  if (tid >= NBATCH * Cout) return;
  int o = tid % Cout;
  int b = tid / Cout;
  const float* xr = x + (size_t)b * Cin;
  const float* wr = w + (size_t)o * Cin;
  float acc = bias[o];
  for (int c = 0; c < Cin; ++c) acc += wr[c] * xr[c];
  y[tid] = act ? fmaxf(acc, 0.f) : acc;
}

// ft' [b,d,n] = sum_c ft[b,c,n] * (tsfm[b,c,d] + I)
__global__ void transform_kernel(const float* __restrict__ ft, const float* __restrict__ tl,
                                 float* __restrict__ out) {
  int tid = blockIdx.x * blockDim.x + threadIdx.x;
  int total = NBATCH * FD * NSAMP;
  if (tid >= total) return;
  int n = tid % NSAMP;
  int d = (tid / NSAMP) % FD;
  int b = tid / (NSAMP * FD);
  const float* ftb = ft + (size_t)b * FD * NSAMP + n;
  const float* tb  = tl + (size_t)b * FD * FD + d;
  float acc = 0.f;
  for (int c = 0; c < FD; ++c) {
    float tv = tb[(size_t)c * FD] + ((c == d) ? 1.f : 0.f);
    acc += ftb[(size_t)c * NSAMP] * tv;
  }
  out[tid] = acc;
}

__global__ __launch_bounds__(256) void wpool_kernel(const float* __restrict__ x,
                                                    const float* __restrict__ mass,
                                                    float* __restrict__ y) {
  int bo = blockIdx.x;                 // b*NFEAT + o
  int b = bo / NFEAT;
  int t = threadIdx.x;
  const float* xr = x + (size_t)bo * NSAMP;
  const float* mr = mass + (size_t)b * NSAMP;
  float ws = 0.f, ms = 0.f;
  for (int n = t; n < NSAMP; n += 256) { float m = mr[n]; ws += xr[n] * m; ms += m; }
  __shared__ float sw[256], sm[256];
  sw[t] = ws; sm[t] = ms; __syncthreads();
  for (int off = 128; off > 0; off >>= 1) {
    if (t < off) { sw[t] += sw[t + off]; sm[t] += sm[t + off]; }
    __syncthreads();
  }
  if (t == 0) y[bo] = sw[0] / sm[0];
}

__global__ __launch_bounds__(128) void head_kernel(const float* __restrict__ pooled,
                                                   const float* __restrict__ w0,
                                                   const float* __restrict__ b0,
                                                   const float* __restrict__ w1,
                                                   const float* __restrict__ b1,
                                                   float* __restrict__ out) {
  int b = blockIdx.x;
  int t = threadIdx.x;
  __shared__ float h[NFEAT];
  const float* p = pooled + (size_t)b * NFEAT;
  float acc = b0[t];
  for (int c = 0; c < NFEAT; ++c) acc += w0[(size_t)t * NFEAT + c] * p[c];
  h[t] = fmaxf(acc, 0.f);
  __syncthreads();
  if (t < NLVL) {
    float a2 = b1[t];
    for (int j = 0; j < NFEAT; ++j) a2 += w1[(size_t)t * NFEAT + j] * h[j];
    out[b * NLVL + t] = a2;
  }
}

// ---------------------------------------------------------------------------
extern "C" void kernel_launch(void* const* d_in, const int* in_sizes, int n_in,
                              void* d_out, int out_size, void* d_ws, size_t ws_size,
                              hipStream_t stream) {
  (void)in_sizes; (void)n_in; (void)out_size; (void)ws_size;
  const float* xyz0   = (const float*)d_in[0];
  const float* xyz1   = (const float*)d_in[1];
  const float* evecs0 = (const float*)d_in[2];
  const float* evecs1 = (const float*)d_in[3];
  const float* mass1  = (const float*)d_in[7];
  const float* fm[NLVL] = {(const float*)d_in[10], (const float*)d_in[11],
                           (const float*)d_in[12], (const float*)d_in[13]};
  // params leaves: JAX pytree flatten (outer dict keys sorted, inner keys sorted)
  const float* fc0_b  = (const float*)d_in[14];
  const float* fc0_be = (const float*)d_in[15];
  const float* fc0_g  = (const float*)d_in[16];
  const float* fc0_m  = (const float*)d_in[17];
  const float* fc0_v  = (const float*)d_in[18];
  const float* fc0_w  = (const float*)d_in[19];
  const float* fc1_b  = (const float*)d_in[20];
  const float* fc1_be = (const float*)d_in[21];
  const float* fc1_g  = (const float*)d_in[22];
  const float* fc1_m  = (const float*)d_in[23];
  const float* fc1_v  = (const float*)d_in[24];
  const float* fc1_w  = (const float*)d_in[25];
  const float* fl0_b  = (const float*)d_in[26];
  const float* fl0_w  = (const float*)d_in[27];
  const float* fl1_b  = (const float*)d_in[28];
  const float* fl1_w  = (const float*)d_in[29];
  const float* m0_b   = (const float*)d_in[30];
  const float* m0_be  = (const float*)d_in[31];
  const float* m0_g   = (const float*)d_in[32];
  const float* m0_m   = (const float*)d_in[33];
  const float* m0_v   = (const float*)d_in[34];
  const float* m0_w   = (const float*)d_in[35];
  const float* m10_b  = (const float*)d_in[36];
  const float* m10_be = (const float*)d_in[37];
  const float* m10_g  = (const float*)d_in[38];
  const float* m10_m  = (const float*)d_in[39];
  const float* m10_v  = (const float*)d_in[40];
  const float* m10_w  = (const float*)d_in[41];
  const float* m11_b  = (const float*)d_in[42];
  const float* m11_be = (const float*)d_in[43];
  const float* m11_g  = (const float*)d_in[44];
  const float* m11_m  = (const float*)d_in[45];
  const float* m11_v  = (const float*)d_in[46];
  const float* m11_w  = (const float*)d_in[47];
  const float* l0_b   = (const float*)d_in[48];
  const float* l0_w   = (const float*)d_in[49];
  const float* l1_b   = (const float*)d_in[50];
  const float* l1_w   = (const float*)d_in[51];
  float* out = (float*)d_out;

  // workspace carve-out
  char* ws = (char*)d_ws;
  size_t off = 0;
  auto carve = [&](size_t bytes) -> char* {
    char* p = ws + off;
    off += (bytes + 255) & ~(size_t)255;
    return p;
  };
  const int K1[NLVL]  = {50, 100, 150, 200};
  const int K1P[NLVL] = {52, 100, 152, 200};   // padded to multiple of 4

  int*   idx0   = (int*)carve(sizeof(int) * NBATCH * NSAMP);
  int*   idx1   = (int*)carve(sizeof(int) * NBATCH * NSAMP);
  float* ev0s   = (float*)carve(sizeof(float) * NBATCH * NSAMP * KMAX);
  float* ev1s   = (float*)carve(sizeof(float) * NBATCH * NSAMP * KMAX);
  float* mass1s = (float*)carve(sizeof(float) * NBATCH * NSAMP);
  float* e0all  = (float*)carve(sizeof(float) * NBATCH * NSAMP * (52 + 100 + 152 + 200));
  float* n0all  = (float*)carve(sizeof(float) * NLVL * NBATCH * NSAMP);
  float* n1all  = (float*)carve(sizeof(float) * NLVL * NBATCH * NSAMP);
  float* res    = (float*)carve(sizeof(float) * NBATCH * NLVL * NSAMP);
  float* ft     = (float*)carve(sizeof(float) * NBATCH * FD * NSAMP);
  float* y1     = (float*)carve(sizeof(float) * NBATCH * FD * NSAMP);
  float* y2     = (float*)carve(sizeof(float) * NBATCH * FD * NSAMP);
  float* gmax   = (float*)carve(sizeof(float) * NBATCH * FD);
  float* fh     = (float*)carve(sizeof(float) * NBATCH * FD);
  float* tsf    = (float*)carve(sizeof(float) * NBATCH * FD * FD);
  float* ftT    = (float*)carve(sizeof(float) * NBATCH * FD * NSAMP);
  float* m1a    = (float*)carve(sizeof(float) * NBATCH * NFEAT * NSAMP);
  float* m1b    = (float*)carve(sizeof(float) * NBATCH * NFEAT * NSAMP);
  float* pooled = (float*)carve(sizeof(float) * NBATCH * NFEAT);

  // 1) FPS (latency-dominant sequential chains)
  fps_kernel<<<dim3(NBATCH), dim3(1024), 0, stream>>>(xyz0, idx0);
  fps_kernel<<<dim3(NBATCH), dim3(1024), 0, stream>>>(xyz1, idx1);

  // 2) gather
  {
    int total = NBATCH * NSAMP * KMAX;
    gather_kernel<<<(total + 255) / 256, 256, 0, stream>>>(
        evecs0, evecs1, mass1, idx0, idx1, ev0s, ev1s, mass1s);
  }

  // 3) residuals per spectral level (WMMA distance GEMM)
  size_t e0off = 0;
  for (int l = 0; l < NLVL; ++l) {
    int k1 = K1[l], k1p = K1P[l];
    float* e0 = e0all + e0off;
    float* n0 = n0all + (size_t)l * NBATCH * NSAMP;
    float* n1 = n1all + (size_t)l * NBATCH * NSAMP;
    {
      int total = NBATCH * NSAMP * k1p;
      e0_gemm_kernel<<<(total + 255) / 256, 256, 0, stream>>>(ev0s, fm[l], e0, k1, k1p);
    }
    norms_kernel<<<(NBATCH * NSAMP + 255) / 256, 256, 0, stream>>>(ev1s, e0, n1, n0, k1, k1p);
    float scale = 1.0f / sqrtf((float)k1);
    dim3 grid(NSAMP / 16, NBATCH), blk(256);
    switch (l) {
      case 0: pd_min_wmma_kernel< 50,  52><<<grid, blk, 0, stream>>>(ev1s, e0, n1, n0, res, scale, l); break;
      case 1: pd_min_wmma_kernel<100, 100><<<grid, blk, 0, stream>>>(ev1s, e0, n1, n0, res, scale, l); break;
      case 2: pd_min_wmma_kernel<150, 152><<<grid, blk, 0, stream>>>(ev1s, e0, n1, n0, res, scale, l); break;
      default: pd_min_wmma_kernel<200, 200><<<grid, blk, 0, stream>>>(ev1s, e0, n1, n0, res, scale, l); break;
    }
    e0off += (size_t)NBATCH * NSAMP * k1p;
  }

  // 4) network head
  conv_bn_kernel<<<(NBATCH * FD * NSAMP + 255) / 256, 256, 0, stream>>>(
      res, m0_w, m0_b, m0_g, m0_be, m0_m, m0_v, ft, NLVL, FD, 1);
  conv_bn_kernel<<<(NBATCH * FD * NSAMP + 255) / 256, 256, 0, stream>>>(
      ft, fc0_w, fc0_b, fc0_g, fc0_be, fc0_m, fc0_v, y1, FD, FD, 1);
  conv_bn_kernel<<<(NBATCH * FD * NSAMP + 255) / 256, 256, 0, stream>>>(
      y1, fc1_w, fc1_b, fc1_g, fc1_be, fc1_m, fc1_v, y2, FD, FD, 1);
  maxpool_kernel<<<NBATCH * FD, 256, 0, stream>>>(y2, gmax);
  linear_kernel<<<1, 256, 0, stream>>>(gmax, fl0_w, fl0_b, fh, FD, FD, 1);
  linear_kernel<<<(NBATCH * FD * FD + 255) / 256, 256, 0, stream>>>(
      fh, fl1_w, fl1_b, tsf, FD, FD * FD, 0);
  transform_kernel<<<(NBATCH * FD * NSAMP + 255) / 256, 256, 0, stream>>>(ft, tsf, ftT);
  conv_bn_kernel<<<(NBATCH * NFEAT * NSAMP + 255) / 256, 256, 0, stream>>>(
      ftT, m10_w, m10_b, m10_g, m10_be, m10_m, m10_v, m1a, FD, NFEAT, 1);
  conv_bn_kernel<<<(NBATCH * NFEAT * NSAMP + 255) / 256, 256, 0, stream>>>(
      m1a, m11_w, m11_b, m11_g, m11_be, m11_m, m11_v, m1b, NFEAT, NFEAT, 0);
  wpool_kernel<<<NBATCH * NFEAT, 256, 0, stream>>>(m1b, mass1s, pooled);
  head_kernel<<<NBATCH, NFEAT, 0, stream>>>(pooled, l0_w, l0_b, l1_w, l1_b, out);
}